// TorchQuantumCircuit_76794015252978
// MI455X (gfx1250) — compile-verified
//
#include <hip/hip_runtime.h>
#include <math.h>

// ---------------------------------------------------------------------------
// 12-qubit state-vector circuit on MI455X (gfx1250):
//  - one workgroup (256 thr = 8 wave32) per batch element, state in LDS
//  - whole circuit expressed as 36 applications of 16x16 complex matrices on
//    4-bit groups; each apply done with V_WMMA_F32_16X16X4_F32 (K chained x4),
//    complex mult via 3 accumulators (P=Ar*Br, Q=Ai*Bi, I=Ar*Bi+Ai*Br).
//  - WMMA layouts per CDNA5 ISA 7.12.2:
//      A 16x4 f32 : lane m=lane&15 rows; VGPR v holds K = v + 2*(lane>=16)
//      C/D 16x16  : VGPR v holds M = v + 8*(lane>=16), N = lane&15
//      B 4x16     : assumed to mirror A's K-interleave: VGPR v, lane half ->
//                   K = v + 2*(lane>=16), N = lane&15   (not hardware-verified)
//  - all LDS addressing is algebraic (bit-deposit identities), no runtime-
//    indexed private arrays -> no v_cndmask select chains.
// ---------------------------------------------------------------------------

typedef __attribute__((ext_vector_type(2))) float v2f;
typedef __attribute__((ext_vector_type(8))) float v8f;

#define NQ    12
#define DIMS  4096
#define NSLOT 36

// 36 slots: [block b=0..2 (ZZ,XX,YY)] x 12:
//  0..2  fused (encoding * in-group Ising) on groups A/B/C (enc part built in main)
//  3     crossing Ising pairs {3,4} & {7,8} on group (3,4,7,8)
//  4     crossing Ising pair {0,11} on group (0,11,1,2)
//  5..7  bias+trainable Kron on groups A/B/C
//  8..11 CNOT-ring chunks on (8,9,10,11),(5,6,7,8),(2,3,4,5),(0,1,2,11)
__device__ float g_wsmat[NSLOT * 512]; // per slot: 256 real + 256 imag (row-major 16x16)

__constant__ int APPLY_Q[12][4] = {
    {0,1,2,3},{4,5,6,7},{8,9,10,11},
    {3,4,7,8},{0,11,1,2},
    {0,1,2,3},{4,5,6,7},{8,9,10,11},
    {8,9,10,11},{5,6,7,8},{2,3,4,5},{0,1,2,11}
};

struct C2 { float x, y; };
static __device__ __forceinline__ C2 cmul(C2 a, C2 b){ C2 r; r.x=a.x*b.x-a.y*b.y; r.y=a.x*b.y+a.y*b.x; return r; }
static __device__ __forceinline__ C2 cadd(C2 a, C2 b){ C2 r; r.x=a.x+b.x; r.y=a.y+b.y; return r; }

struct M2 { C2 m[2][2]; };
struct M4 { C2 m[4][4]; };

static __device__ __forceinline__ M2 mul2(M2 a, M2 b){
    M2 c;
    for (int i=0;i<2;++i) for (int j=0;j<2;++j){
        C2 s={0.f,0.f};
        for (int k=0;k<2;++k) s = cadd(s, cmul(a.m[i][k], b.m[k][j]));
        c.m[i][j]=s;
    }
    return c;
}
static __device__ __forceinline__ M2 g_rx(float t){
    float c=__cosf(0.5f*t), s=__sinf(0.5f*t);
    M2 r; r.m[0][0]={c,0.f}; r.m[0][1]={0.f,-s}; r.m[1][0]={0.f,-s}; r.m[1][1]={c,0.f}; return r;
}
static __device__ __forceinline__ M2 g_ry(float t){
    float c=__cosf(0.5f*t), s=__sinf(0.5f*t);
    M2 r; r.m[0][0]={c,0.f}; r.m[0][1]={-s,0.f}; r.m[1][0]={s,0.f}; r.m[1][1]={c,0.f}; return r;
}
static __device__ __forceinline__ M2 g_rz(float t){
    float c=__cosf(0.5f*t), s=__sinf(0.5f*t);
    M2 r; r.m[0][0]={c,-s}; r.m[0][1]={0.f,0.f}; r.m[1][0]={0.f,0.f}; r.m[1][1]={c,s}; return r;
}
static __device__ __forceinline__ M4 g_zz(float t){
    float c=__cosf(0.5f*t), s=__sinf(0.5f*t);
    M4 r;
    for (int i=0;i<4;++i) for (int j=0;j<4;++j) r.m[i][j]={0.f,0.f};
    r.m[0][0]={c,-s}; r.m[1][1]={c,s}; r.m[2][2]={c,s}; r.m[3][3]={c,-s};
    return r;
}
static __device__ __forceinline__ M4 g_xx(float t){
    float c=__cosf(0.5f*t), s=__sinf(0.5f*t);
    M4 r;
    for (int i=0;i<4;++i) for (int j=0;j<4;++j) r.m[i][j]={0.f,0.f};
    for (int i=0;i<4;++i){ r.m[i][i]={c,0.f}; r.m[i][3-i]={0.f,-s}; }
    return r;
}
static __device__ __forceinline__ M4 g_yy(float t){
    float c=__cosf(0.5f*t), s=__sinf(0.5f*t);
    M4 r;
    for (int i=0;i<4;++i) for (int j=0;j<4;++j) r.m[i][j]={0.f,0.f};
    for (int i=0;i<4;++i) r.m[i][i]={c,0.f};
    r.m[0][3]={0.f,s}; r.m[3][0]={0.f,s}; r.m[1][2]={0.f,-s}; r.m[2][1]={0.f,-s};
    return r;
}

// element of (g embedded at local bits la<lb, identity elsewhere); gate MSB <-> lb
static __device__ __forceinline__ C2 embVal(const M4& g, int la, int lb, int i, int j){
    int pm = (1<<la)|(1<<lb);
    if (((i ^ j) & ~pm & 15) != 0) return C2{0.f,0.f};
    int ri = 2*((i>>lb)&1) + ((i>>la)&1);
    int cj = 2*((j>>lb)&1) + ((j>>la)&1);
    return g.m[ri][cj];
}
// element (i,j) of Emb(g, la, lb) * B  (B given as 16x16 real/imag planes)
static __device__ __forceinline__ C2 embMulVal(const M4& g, int la, int lb,
                                               const float* Br, const float* Bi,
                                               int i, int j){
    int pm = (1<<la)|(1<<lb);
    int row = 2*((i>>lb)&1) + ((i>>la)&1);
    int base = i & ~pm & 15;
    C2 acc = {0.f,0.f};
    #pragma unroll
    for (int t=0;t<4;++t){
        int k = base | ((t&1)<<la) | (((t>>1)&1)<<lb);
        C2 bv = { Br[k*16+j], Bi[k*16+j] };
        acc = cadd(acc, cmul(g.m[row][t], bv));
    }
    return acc;
}

// -------------------- prep kernel: parameter-only 16x16 matrices ------------
__global__ __launch_bounds__(256)
void qc_prep_kernel(const float* eta, const float* ew_zz, const float* ew_xx, const float* ew_yy,
                    const float* nb_z, const float* nb_x, const float* nb_y, const float* trots)
{
    __shared__ M4 sE[12];          // Ising 4x4 per ring edge (bit-space)
    __shared__ M2 sT[12];          // bias+trainable 2x2 per bit
    __shared__ float t1r[256], t1i[256], t2r[256], t2i[256];
    int tid = threadIdx.x;
    int i = tid >> 4, j = tid & 15;

    for (int b = 0; b < 3; ++b) {
        if (tid < 12) {
            const float* ew = (b==0)?ew_zz:((b==1)?ew_xx:ew_yy);
            float th = eta[b] * ew[tid];
            sE[tid] = (b==0)?g_zz(th):((b==1)?g_xx(th):g_yy(th));
        } else if (tid >= 16 && tid < 28) {
            int q = tid - 16;
            const float* nb = (b==0)?nb_z:((b==1)?nb_x:nb_y);
            M2 bias = (b==0)?g_rz(nb[q]):((b==1)?g_rx(nb[q]):g_ry(nb[q]));
            const float* rr = trots + (b*12 + q)*3;
            sT[q] = mul2(g_rz(rr[2]), mul2(g_ry(rr[1]), mul2(g_rx(rr[0]), bias)));
        }
        __syncthreads();

        // in-group Ising: group gi covers bits {4gi..4gi+3}; local pair (0,1)->edge 10-4gi, (1,2)->9-4gi, (2,3)->8-4gi
        for (int gi = 0; gi < 3; ++gi) {
            C2 v = embVal(sE[10-4*gi], 0, 1, i, j);
            t1r[tid]=v.x; t1i[tid]=v.y;
            __syncthreads();
            v = embMulVal(sE[9-4*gi], 1, 2, t1r, t1i, i, j);
            t2r[tid]=v.x; t2i[tid]=v.y;
            __syncthreads();
            v = embMulVal(sE[8-4*gi], 2, 3, t2r, t2i, i, j);
            float* w = g_wsmat + (b*12+gi)*512;
            w[tid]=v.x; w[256+tid]=v.y;
            __syncthreads();
        }
        // crossing pair {3,4}(edge7) at locals(0,1) and {7,8}(edge3) at locals(2,3) of group (3,4,7,8)
        {
            C2 v = embVal(sE[7], 0, 1, i, j);
            t1r[tid]=v.x; t1i[tid]=v.y;
            __syncthreads();
            v = embMulVal(sE[3], 2, 3, t1r, t1i, i, j);
            float* w = g_wsmat + (b*12+3)*512;
            w[tid]=v.x; w[256+tid]=v.y;
            __syncthreads();
        }
        // crossing pair {0,11}(edge11) at locals(0,1) of group (0,11,1,2)
        {
            C2 v = embVal(sE[11], 0, 1, i, j);
            float* w = g_wsmat + (b*12+4)*512;
            w[tid]=v.x; w[256+tid]=v.y;
        }
        // bias+trainable Kron: slots 5..7
        for (int gi = 0; gi < 3; ++gi) {
            C2 v = sT[4*gi+0].m[i&1][j&1];
            v = cmul(v, sT[4*gi+1].m[(i>>1)&1][(j>>1)&1]);
            v = cmul(v, sT[4*gi+2].m[(i>>2)&1][(j>>2)&1]);
            v = cmul(v, sT[4*gi+3].m[(i>>3)&1][(j>>3)&1]);
            float* w = g_wsmat + (b*12+5+gi)*512;
            w[tid]=v.x; w[256+tid]=v.y;
        }
        // CNOT chunks: permutation matrices; M[i][j] = [i == cc(cb(ca(j)))]
        for (int ch = 0; ch < 4; ++ch) {
            int f = j;
            if (ch < 3) {
                f ^= ((f>>3)&1)<<2;   // C(l3->l2)
                f ^= ((f>>2)&1)<<1;   // C(l2->l1)
                f ^= ((f>>1)&1);      // C(l1->l0)
            } else {
                f ^= ((f>>2)&1)<<1;   // C(l2->l1)
                f ^= ((f>>1)&1);      // C(l1->l0)
                f ^= ((f>>3)&1);      // C(l3->l0)
            }
            float* w = g_wsmat + (b*12+8+ch)*512;
            w[tid] = (i==f) ? 1.0f : 0.0f;
            w[256+tid] = 0.0f;
        }
        __syncthreads();
    }
}

// -------------------- WMMA helpers ------------------------------------------
static __device__ __forceinline__ v8f wmma4(v2f a, v2f b, v8f c){
    // D = A(16x4 f32) x B(4x16 f32) + C(16x16 f32)
    return __builtin_amdgcn_wmma_f32_16x16x4_f32(false, a, false, b, (short)0, c, false, false);
}
// insert zero bits at sorted-ascending positions p0<p1<p2<p3 (bit-deposit perm)
static __device__ __forceinline__ int bitSpread(int x, int p0, int p1, int p2, int p3){
    x = ((x >> p0) << (p0+1)) | (x & ((1<<p0)-1));
    x = ((x >> p1) << (p1+1)) | (x & ((1<<p1)-1));
    x = ((x >> p2) << (p2+1)) | (x & ((1<<p2)-1));
    x = ((x >> p3) << (p3+1)) | (x & ((1<<p3)-1));
    return x;
}
static __device__ __forceinline__ void cswap(int& a, int& b){ if (a > b){ int t=a; a=b; b=t; } }

// -------------------- main kernel: one workgroup per batch element ----------
__global__ __launch_bounds__(256)
void qc_main_kernel(const float* data, const float* escale, const float* ebias, float* out)
{
    __shared__ float sr[DIMS], si[DIMS];     // state, real/imag
    __shared__ float gr[256], gi[256];       // current 16x16 gate (row-major)
    __shared__ float kr[256], ki[256];       // encoding Kron scratch
    __shared__ M2 sEg[4];
    __shared__ float red[256];

    int tid = threadIdx.x, bat = blockIdx.x;
    int lane = tid & 31, wave = tid >> 5;
    int hi = lane >> 4, m = lane & 15;

    for (int k = tid; k < DIMS; k += 256){ sr[k]=0.f; si[k]=0.f; }
    __syncthreads();
    if (tid == 0) sr[0] = 1.f;
    __syncthreads();

    for (int slot = 0; slot < NSLOT; ++slot) {
        int b = slot / 12, loc = slot % 12;
        int q0 = APPLY_Q[loc][0], q1 = APPLY_Q[loc][1], q2 = APPLY_Q[loc][2], q3 = APPLY_Q[loc][3];
        const float* wmr = g_wsmat + slot*512;
        const float* wmi = wmr + 256;

        if (loc < 3) {
            // batch-dependent encoding gates, fused with in-group Ising: G = IsingIn * Kron(E)
            if (tid < 4) {
                int q = APPLY_Q[loc][tid];
                float d  = data[bat*12 + q];
                float ax = escale[q*3+0]*d + ebias[q*3+0];
                float ay = escale[q*3+1]*d + ebias[q*3+1];
                float az = escale[q*3+2]*d + ebias[q*3+2];
                M2 e;
                if (b == 0)      e = mul2(g_rz(az), mul2(g_ry(ay), g_rx(ax))); // full
                else if (b == 1) e = mul2(g_ry(ay), g_rx(ax));                 // xy
                else             e = mul2(g_rz(az), g_ry(ay));                 // yz
                sEg[tid] = e;
            }
            __syncthreads();
            {
                int i = tid >> 4, j = tid & 15;
                C2 v = sEg[0].m[i&1][j&1];
                v = cmul(v, sEg[1].m[(i>>1)&1][(j>>1)&1]);
                v = cmul(v, sEg[2].m[(i>>2)&1][(j>>2)&1]);
                v = cmul(v, sEg[3].m[(i>>3)&1][(j>>3)&1]);
                kr[tid]=v.x; ki[tid]=v.y;
            }
            __syncthreads();
            {
                int i = tid >> 4, j = tid & 15;
                C2 acc = {0.f,0.f};
                #pragma unroll
                for (int k=0;k<16;++k){
                    C2 w = { wmr[i*16+k], wmi[i*16+k] };
                    C2 e = { kr[k*16+j], ki[k*16+j] };
                    acc = cadd(acc, cmul(w,e));
                }
                gr[tid]=acc.x; gi[tid]=acc.y;
            }
        } else {
            gr[tid] = wmr[tid];
            gi[tid] = wmi[tid];
        }
        __syncthreads();

        // sorted group bits for address spreading
        int s0=q0, s1=q1, s2=q2, s3=q3;
        cswap(s0,s1); cswap(s2,s3); cswap(s0,s2); cswap(s1,s3); cswap(s1,s2);
        // deposit targets of residual bits 0..3 (uniform per apply)
        int D0 = bitSpread(1, s0,s1,s2,s3);
        int D1 = bitSpread(2, s0,s1,s2,s3);
        int D2 = bitSpread(4, s0,s1,s2,s3);
        int D3 = bitSpread(8, s0,s1,s2,s3);
        // per-lane pieces
        int jofm = ((m&1)<<q0) | (((m>>1)&1)<<q1) | (((m>>2)&1)<<q2) | (((m>>3)&1)<<q3);
        int spm  = ((m&1)?D0:0) | ((m&2)?D1:0) | ((m&4)?D2:0) | ((m&8)?D3:0);
        int hD3  = hi ? D3 : 0;
        int hq1  = hi << q1;          // bit1 of jj = hi  (jj = 4t + 2*hi)
        int b0q  = 1 << q0;           // jof[jj+1] = jof[jj] + (1<<q0)

        // B operands (gate), loop-invariant across chunks: B[k][n] = U[n][4t+k]
        v2f Br[4], Bi[4];
        #pragma unroll
        for (int t=0;t<4;++t){
            int c0 = 4*t + 2*hi;
            Br[t].x = gr[m*16 + c0]; Br[t].y = gr[m*16 + c0 + 1];
            Bi[t].x = gi[m*16 + c0]; Bi[t].y = gi[m*16 + c0 + 1];
        }

        #pragma unroll
        for (int cc = 0; cc < 2; ++cc) {
            int gb = (wave*2 + cc) << 4;
            int spBase = bitSpread(gb, s0, s1, s2, s3);   // uniform per chunk
            int rdBase = spBase + spm + hq1;              // read rows: group gb+m
            int wrBase = spBase + hD3 + jofm;             // write rows: gb + v + 8*hi
            v8f accP = {0.f,0.f,0.f,0.f,0.f,0.f,0.f,0.f};
            v8f accQ = {0.f,0.f,0.f,0.f,0.f,0.f,0.f,0.f};
            v8f accI = {0.f,0.f,0.f,0.f,0.f,0.f,0.f,0.f};
            #pragma unroll
            for (int t=0;t<4;++t){
                int a0 = rdBase + ((t&1)<<q2) + ((t>>1)<<q3);
                int a1 = a0 + b0q;
                v2f ar, ai;
                ar.x = sr[a0]; ar.y = sr[a1];
                ai.x = si[a0]; ai.y = si[a1];
                accP = wmma4(ar, Br[t], accP);   // Ar*Br
                accQ = wmma4(ai, Bi[t], accQ);   // Ai*Bi
                accI = wmma4(ar, Bi[t], accI);   // Ar*Bi
                accI = wmma4(ai, Br[t], accI);   // + Ai*Br
            }
            v8f dr = accP - accQ;
            #pragma unroll
            for (int v=0; v<8; ++v){
                int sv = ((v&1)?D0:0) + ((v&2)?D1:0) + ((v&4)?D2:0);  // v compile-time: folds
                int adr = wrBase + sv;
                sr[adr] = dr[v];
                si[adr] = accI[v];
            }
        }
        __syncthreads();
    }

    // Pauli-Z expectations: out[bat,i] = sum_k |s_k|^2 * (1 - 2*bit_{11-i}(k))
    float acc[12];
    #pragma unroll
    for (int i=0;i<12;++i) acc[i]=0.f;
    for (int k = tid; k < DIMS; k += 256){
        float p = sr[k]*sr[k] + si[k]*si[k];
        #pragma unroll
        for (int i=0;i<12;++i) acc[i] += ((k >> (11-i)) & 1) ? -p : p;
    }
    for (int i=0;i<12;++i){
        __syncthreads();
        red[tid] = acc[i];
        __syncthreads();
        for (int s=128; s>0; s>>=1){
            if (tid < s) red[tid] += red[tid+s];
            __syncthreads();
        }
        if (tid == 0) out[bat*12 + i] = red[0];
    }
}

extern "C" void kernel_launch(void* const* d_in, const int* in_sizes, int n_in,
                              void* d_out, int out_size, void* d_ws, size_t ws_size,
                              hipStream_t stream) {
    (void)in_sizes; (void)n_in; (void)out_size; (void)d_ws; (void)ws_size;
    const float* data   = (const float*)d_in[0];
    const float* escale = (const float*)d_in[1];
    const float* ebias  = (const float*)d_in[2];
    const float* eta    = (const float*)d_in[3];
    const float* ew_zz  = (const float*)d_in[4];
    const float* ew_xx  = (const float*)d_in[5];
    const float* ew_yy  = (const float*)d_in[6];
    const float* nb_z   = (const float*)d_in[7];
    const float* nb_x   = (const float*)d_in[8];
    const float* nb_y   = (const float*)d_in[9];
    const float* trots  = (const float*)d_in[10];
    float* out = (float*)d_out;

    qc_prep_kernel<<<1, 256, 0, stream>>>(eta, ew_zz, ew_xx, ew_yy, nb_z, nb_x, nb_y, trots);
    qc_main_kernel<<<64, 256, 0, stream>>>(data, escale, ebias, out);
}